// NeuralODETorch_86406152061713
// MI455X (gfx1250) — compile-verified
//
#include <hip/hip_runtime.h>

typedef __attribute__((ext_vector_type(2))) float v2f;
typedef __attribute__((ext_vector_type(8))) float v8f;

#define WAVES_PER_BLOCK 8
#define TILE_M 16
constexpr int BATCH    = 131072;
constexpr int NSAVE    = 100;
constexpr int SUBSTEPS = 4;   // RK4 substeps per save interval; h ~ 2.5e-3 -> err << 1e-6

#define LOG2E_F  1.44269504088896340736f
#define LN2_F    0.69314718055994530942f

// softplus(x) = max(x,0) + ln2 * log2(1 + 2^(-log2e*|x|))
// exp2 arg <= 0: no overflow (underflow -> 0 -> log2(1)=0, exact).
// log2 arg in [1,2]: no denormals, native v_log_f32 is accurate there.
__device__ __forceinline__ float softplus_f(float x) {
    float e = __builtin_amdgcn_exp2f(-LOG2E_F * fabsf(x));
    return fmaxf(x, 0.0f) + LN2_F * __builtin_amdgcn_logf(1.0f + e);
}

// tanh: prefer CDNA5 hardware v_tanh_f32; fallback 1 - 2*rcp(1+e^{2x})
// saturates correctly at +/-1 (exp2 -> inf -> rcp -> 0; exp2 -> 0 -> -1).
__device__ __forceinline__ float tanh_f(float x) {
#if defined(__has_builtin) && __has_builtin(__builtin_amdgcn_tanhf)
    return __builtin_amdgcn_tanhf(x);
#else
    float e = __builtin_amdgcn_exp2f((2.0f * LOG2E_F) * x);
    return fmaf(-2.0f, __builtin_amdgcn_rcpf(1.0f + e), 1.0f);
#endif
}

__global__ void __launch_bounds__(256)
node_rk4_kernel(const float* __restrict__ y0, const float* __restrict__ W1,
                const float* __restrict__ b1, const float* __restrict__ W2,
                const float* __restrict__ b2, const float* __restrict__ t1p,
                float* __restrict__ partial)
{
    // per-wave private LDS: H tile (16x32 f32) + D tile (16x16 f32)
    __shared__ float lds[WAVES_PER_BLOCK * (16 * 32 + 16 * 16)];
    __shared__ float redbuf[WAVES_PER_BLOCK];

    const int lane = threadIdx.x & 31;
    const int wave = threadIdx.x >> 5;
    const int half = lane >> 4;    // 0: lanes 0-15, 1: lanes 16-31
    const int l16  = lane & 15;
    const int m    = l16;          // A row / C column index handled by this lane
    const int k0   = half * 2;     // A-layout: VGPR0 holds K=k0, VGPR1 holds K=k0+1

    float* Hlds = lds + wave * (16 * 32 + 16 * 16);
    float* Dlds = Hlds + 16 * 32;

    const int tile = blockIdx.x * WAVES_PER_BLOCK + wave;   // 0..8191
    const int m0   = tile * TILE_M;

    // ---- load state tile in WMMA A-layout (16x4 f32 A: 2 VGPRs/lane) ----
    v2f y;
    {
        const float* p = y0 + (size_t)(m0 + m) * 4 + k0;
        y.x = p[0];
        y.y = p[1];
    }

    // ---- W1^T as two 4x16 B-layout tiles (B1[k][n] = W1[n][k]) ----
    v2f B1t0, B1t1;
    B1t0.x = W1[(l16)      * 4 + k0];
    B1t0.y = W1[(l16)      * 4 + k0 + 1];
    B1t1.x = W1[(l16 + 16) * 4 + k0];
    B1t1.y = W1[(l16 + 16) * 4 + k0 + 1];
    const float bias1_t0 = b1[l16];
    const float bias1_t1 = b1[l16 + 16];

    // ---- W2^T (32x4) as eight 4x16 B-layout chunks, N>=4 zero-padded ----
    const bool ncol = (l16 < 4);
    v2f B2[8];
#pragma unroll
    for (int c = 0; c < 8; ++c) {
        const int k = 4 * c + k0;
        B2[c].x = ncol ? W2[l16 * 32 + k]     : 0.0f;
        B2[c].y = ncol ? W2[l16 * 32 + k + 1] : 0.0f;
    }
    const float bias2 = ncol ? b2[l16] : 0.0f;
    const float t1    = t1p[0];

    // ---- vector field: dy = tanh(softplus(y W1^T + b1) W2^T + b2) ----
    auto feval = [&](v2f yy) -> v2f {
        // layer 1: two f32 WMMAs, bias pre-loaded into C
        v8f c0, c1;
#pragma unroll
        for (int i = 0; i < 8; ++i) { c0[i] = bias1_t0; c1[i] = bias1_t1; }
        v8f h0 = __builtin_amdgcn_wmma_f32_16x16x4_f32(false, yy, false, B1t0,
                                                       (short)0, c0, false, false);
        v8f h1 = __builtin_amdgcn_wmma_f32_16x16x4_f32(false, yy, false, B1t1,
                                                       (short)0, c1, false, false);
#pragma unroll
        for (int i = 0; i < 8; ++i) { h0[i] = softplus_f(h0[i]); h1[i] = softplus_f(h1[i]); }

        // C-layout -> A-layout transpose of H through wave-private LDS
#pragma unroll
        for (int r = 0; r < 8; ++r) {
            const int mm = r + half * 8;
            Hlds[mm * 32 + l16]      = h0[r];
            Hlds[mm * 32 + l16 + 16] = h1[r];
        }

        // layer 2: chain 8 accumulating WMMAs over K=32
        v8f d2;
#pragma unroll
        for (int i = 0; i < 8; ++i) d2[i] = bias2;
#pragma unroll
        for (int c = 0; c < 8; ++c) {
            v2f a2;
            a2.x = Hlds[m * 32 + 4 * c + k0];
            a2.y = Hlds[m * 32 + 4 * c + k0 + 1];
            d2 = __builtin_amdgcn_wmma_f32_16x16x4_f32(false, a2, false, B2[c],
                                                       (short)0, d2, false, false);
        }

        // tanh, then C-layout -> A-layout dy via LDS
#pragma unroll
        for (int r = 0; r < 8; ++r) {
            const int mm = r + half * 8;
            Dlds[mm * 16 + l16] = tanh_f(d2[r]);
        }
        v2f dy;
        dy.x = Dlds[m * 16 + k0];
        dy.y = Dlds[m * 16 + k0 + 1];
        return dy;
    };

    // ---- integrate; accumulate sum over all save points ----
    float sumacc = y.x + y.y;   // t = 0 save point

    const float dt_save = t1 / (float)(NSAVE - 1);
    const float h       = dt_save / (float)SUBSTEPS;

#pragma unroll 1
    for (int j = 1; j < NSAVE; ++j) {
#pragma unroll 1
        for (int s = 0; s < SUBSTEPS; ++s) {
            v2f k1 = feval(y);
            v2f yt;
            yt.x = y.x + 0.5f * h * k1.x;  yt.y = y.y + 0.5f * h * k1.y;
            v2f k2 = feval(yt);
            yt.x = y.x + 0.5f * h * k2.x;  yt.y = y.y + 0.5f * h * k2.y;
            v2f k3 = feval(yt);
            yt.x = y.x + h * k3.x;         yt.y = y.y + h * k3.y;
            v2f k4 = feval(yt);
            y.x += (h / 6.0f) * (k1.x + 2.0f * k2.x + 2.0f * k3.x + k4.x);
            y.y += (h / 6.0f) * (k1.y + 2.0f * k2.y + 2.0f * k3.y + k4.y);
        }
        sumacc += y.x + y.y;
    }

    // ---- deterministic reduction: wave shuffle -> LDS -> per-block partial ----
#pragma unroll
    for (int off = 16; off >= 1; off >>= 1)
        sumacc += __shfl_xor(sumacc, off, 32);
    if (lane == 0) redbuf[wave] = sumacc;
    __syncthreads();
    if (threadIdx.x == 0) {
        float s = 0.0f;
#pragma unroll
        for (int w = 0; w < WAVES_PER_BLOCK; ++w) s += redbuf[w];
        partial[blockIdx.x] = s;
    }
}

__global__ void __launch_bounds__(256)
node_reduce_kernel(const float* __restrict__ partial, int n, float* __restrict__ out)
{
    __shared__ float buf[256];
    float s = 0.0f;
    for (int i = threadIdx.x; i < n; i += 256) s += partial[i];
    buf[threadIdx.x] = s;
    __syncthreads();
    for (int stride = 128; stride > 0; stride >>= 1) {
        if ((int)threadIdx.x < stride) buf[threadIdx.x] += buf[threadIdx.x + stride];
        __syncthreads();
    }
    if (threadIdx.x == 0) out[0] = buf[0];
}

extern "C" void kernel_launch(void* const* d_in, const int* in_sizes, int n_in,
                              void* d_out, int out_size, void* d_ws, size_t ws_size,
                              hipStream_t stream) {
    (void)in_sizes; (void)n_in; (void)out_size; (void)ws_size;
    const float* y0 = (const float*)d_in[0];
    const float* W1 = (const float*)d_in[1];
    const float* b1 = (const float*)d_in[2];
    const float* W2 = (const float*)d_in[3];
    const float* b2 = (const float*)d_in[4];
    const float* t1 = (const float*)d_in[5];

    float* partial = (float*)d_ws;                 // 1024 floats of scratch
    const int tiles  = BATCH / TILE_M;             // 8192 waves
    const int blocks = tiles / WAVES_PER_BLOCK;    // 1024 blocks of 8 waves

    node_rk4_kernel<<<blocks, 256, 0, stream>>>(y0, W1, b1, W2, b2, t1, partial);
    node_reduce_kernel<<<1, 256, 0, stream>>>(partial, blocks, (float*)d_out);
}